// Sparse_Conv_84731114815958
// MI455X (gfx1250) — compile-verified
//
#include <hip/hip_runtime.h>

// CDNA5 / gfx1250 sparse banded conv via V_WMMA_F32_16X16X4_F32.
// D = A x B + C with A = band-expanded weights (16 out-ch x K=288),
// B = staged X tile (K x 16 pixels), accumulated fully in f32.

typedef __attribute__((ext_vector_type(2))) float v2f;
typedef __attribute__((ext_vector_type(8))) float v8f;

#define IN_CH    128
#define SUB      16
#define XS_CH    32           // 31 real band channels + 1 zero pad channel
#define XS_ROWS  10           // 8 output rows + 2 halo
#define XS_COLS  18           // 16 output cols + 2 halo
#define XS_CHS   (XS_ROWS*XS_COLS)   // 180
#define NCHUNK   72           // K = 288 = 9 taps * 32 dc, in steps of 4
#define A_ELEMS  (NCHUNK*64)  // [chunk][m(16)][sub(4)]

__global__ __launch_bounds__(256)
void sparse_conv_wmma_f32(const float* __restrict__ X,
                          const float* __restrict__ Wt,
                          const float* __restrict__ Bias,
                          float* __restrict__ Out)
{
    __shared__ float Xs[XS_CH * XS_CHS];   // 23040 B
    __shared__ float Alds[A_ELEMS];        // 18432 B
    __shared__ float BiasLds[16];

    const int tid = threadIdx.x;
    const int x0  = blockIdx.x * 16;
    const int y0  = blockIdx.y * 8;
    const int nb  = blockIdx.z >> 3;
    const int o0  = (blockIdx.z & 7) * 16;

    // ---- stage X tile: channels (o0+dc)&127, rows y0..y0+9, cols x0..x0+17 ----
    const float* Xb = X + (size_t)nb * IN_CH * 258 * 258;
    for (int i = tid; i < 31 * XS_CHS; i += 256) {
        int dc  = i / XS_CHS;
        int rem = i - dc * XS_CHS;
        int ry  = rem / XS_COLS;
        int cx  = rem - ry * XS_COLS;
        int ch  = (o0 + dc) & (IN_CH - 1);
        Xs[i] = Xb[(size_t)ch * 258 * 258 + (size_t)(y0 + ry) * 258 + (x0 + cx)];
    }
    // zero the K-padding channel (dc == 31) so padded WMMA terms are 0*0
    for (int i = tid; i < XS_CHS; i += 256) Xs[31 * XS_CHS + i] = 0.0f;

    // ---- stage band-expanded A.  k = tap*32 + dc ; A[m][k] = W[o0+m][dc-m][tap]
    // LDS layout [chunk][m][sub] so each lane reads a contiguous b64 pair.
    for (int i = tid; i < A_ELEMS; i += 256) {
        int c   = i >> 6;
        int rem = i & 63;
        int m   = rem >> 2;
        int sub = rem & 3;
        int k   = c * 4 + sub;
        int tap = k >> 5;
        int dc  = k & 31;
        int s   = dc - m;
        float v = 0.0f;
        if ((unsigned)s < 16u)
            v = Wt[((size_t)(o0 + m) * SUB + s) * 9 + tap];
        Alds[i] = v;
    }
    if (tid < 16) BiasLds[tid] = Bias[o0 + tid];

    __syncthreads();

    // ---- per-wave WMMA: 16 out-ch x 16 x-pixels for one output row ----
    const int lane = tid & 31;
    const int wv   = tid >> 5;        // output row within tile (0..7)
    const int hi   = lane >> 4;       // lane half: handles K sub 2,3 (=> dc+2)
    const int n    = lane & 15;       // pixel col (B) == row m (A)

    // B base: Xs[hi*2][wv][n]; per-chunk offsets are compile-time immediates.
    const char*  xsb = (const char*)Xs + ((wv * XS_COLS + n) + hi * 2 * XS_CHS) * 4;
    const float* ap  = Alds + (n * 4 + hi * 2);

    v8f acc = {0.f, 0.f, 0.f, 0.f, 0.f, 0.f, 0.f, 0.f};

#pragma unroll
    for (int c = 0; c < NCHUNK; ++c) {
        const int tap  = c >> 3;                  // compile-time
        const int kh   = tap / 3;
        const int kw   = tap - kh * 3;
        const int dc0  = (c & 7) * 4;
        const int boff = (dc0 * XS_CHS + kh * XS_COLS + kw) * 4;  // <= 20312

        v2f a = *(const v2f*)(ap + c * 64);                       // ds_load_b64, imm
        v2f b;
        b.x = *(const float*)(xsb + boff);                        // ds_load_b32, imm
        b.y = *(const float*)(xsb + boff + XS_CHS * 4);           // ds_load_b32, imm
        acc = __builtin_amdgcn_wmma_f32_16x16x4_f32(false, a, false, b,
                                                    (short)0, acc, false, false);
    }

    // ---- add bias, store (D layout: M = r + hi*8, N = n) ----
    float* outp = Out + (((size_t)nb * IN_CH + (o0 + hi * 8)) << 16)
                      + ((size_t)(y0 + wv) << 8) + (x0 + n);
#pragma unroll
    for (int r = 0; r < 8; ++r) {
        float v = acc[r] + BiasLds[hi * 8 + r];
        __builtin_nontemporal_store(v, outp + ((size_t)r << 16));
    }
}

extern "C" void kernel_launch(void* const* d_in, const int* in_sizes, int n_in,
                              void* d_out, int out_size, void* d_ws, size_t ws_size,
                              hipStream_t stream) {
    const float* X  = (const float*)d_in[0];   // (8,128,258,258) f32
    const float* W  = (const float*)d_in[1];   // (128,1,16,3,3)  f32
    const float* B  = (const float*)d_in[2];   // (128,1)         f32
    float*       O  = (float*)d_out;           // (8,128,256,256) f32

    dim3 block(256);
    dim3 grid(16 /*x tiles*/, 32 /*y tiles*/, 8 * 8 /*batch*o-tile*/);
    hipLaunchKernelGGL(sparse_conv_wmma_f32, grid, block, 0, stream, X, W, B, O);
}